// GroupLasso_70935679861371
// MI455X (gfx1250) — compile-verified
//
#include <hip/hip_runtime.h>
#include <hip/hip_bf16.h>
#include <math.h>

typedef float v2f __attribute__((ext_vector_type(2)));
typedef float v8f __attribute__((ext_vector_type(8)));

#define D_DIM 128
#define K_DIM 256
#define T_DIM 1024
#define B_DIM 4
#define N_TOT ((size_t)B_DIM * K_DIM * T_DIM)   // 1,048,576
#define LAM_C 0.01f
#define REG_C 0.01f
#define TOL_C 1e-6f
#define MAX_IT 100

// Padded LDS row strides (floats) chosen for bank-conflict-free fragment reads
#define AS_STRIDE 260   // %64 == 4  -> A b64 reads: lanes 0-15 hit banks {4l, 4l+1}
#define BS_STRIDE 272   // %64 == 16 -> B b32 reads: lane halves (rows r, r+2) disjoint banks

// ---------------------------------------------------------------------------
// CDNA5 async global->LDS copy (ASYNCcnt-tracked), per CDNA5 ISA
// ---------------------------------------------------------------------------
__device__ __forceinline__ void async_copy_b128(uint32_t lds_off, const void* gptr) {
  asm volatile("global_load_async_to_lds_b128 %0, %1, off"
               :: "v"(lds_off), "v"((unsigned long long)(uintptr_t)gptr)
               : "memory");
}
__device__ __forceinline__ void wait_async_le4() {
  asm volatile("s_wait_asynccnt 0x4" ::: "memory");
}
__device__ __forceinline__ void wait_async_0() {
  asm volatile("s_wait_asynccnt 0x0" ::: "memory");
}

// ---------------------------------------------------------------------------
// Prologue kernels
// ---------------------------------------------------------------------------

// DtD[i][j] = sum_d D[d][i] * D[d][j]    (D is [128,256] row-major)
__global__ __launch_bounds__(256) void dtd_kernel(const float* __restrict__ D,
                                                  float* __restrict__ DtD) {
  int idx = blockIdx.x * 256 + threadIdx.x;   // 65536 threads
  int i = idx >> 8, j = idx & 255;
  float acc = 0.f;
  for (int d = 0; d < D_DIM; ++d)
    acc += D[d * K_DIM + i] * D[d * K_DIM + j];
  DtD[idx] = acc;
}

// Power iteration for lambda_max(DtD) (PSD -> == max |eig|). Single block.
__global__ __launch_bounds__(256) void power_iter_kernel(const float* __restrict__ DtD,
                                                         float* __restrict__ dLinv) {
  __shared__ float v[K_DIM];
  __shared__ float red[K_DIM];
  const int i = threadIdx.x;
  v[i] = 1.f;
  __syncthreads();
  float nrm = 1.f;
  for (int iter = 0; iter < 128; ++iter) {
    float y = 0.f;
    const float* row = DtD + (size_t)i * K_DIM;
    for (int j = 0; j < K_DIM; ++j) y += row[j] * v[j];
    red[i] = y * y;
    __syncthreads();
    for (int s = 128; s > 0; s >>= 1) {
      if (i < s) red[i] += red[i + s];
      __syncthreads();
    }
    nrm = sqrtf(red[0]);
    __syncthreads();
    v[i] = y / nrm;
    __syncthreads();
  }
  if (i == 0) *dLinv = 1.f / nrm;
}

// A = I - DtD * Linv
__global__ __launch_bounds__(256) void build_A_kernel(const float* __restrict__ DtD,
                                                      const float* __restrict__ dLinv,
                                                      float* __restrict__ A) {
  int idx = blockIdx.x * 256 + threadIdx.x;
  int i = idx >> 8, j = idx & 255;
  float Linv = *dLinv;
  A[idx] = ((i == j) ? 1.f : 0.f) - DtD[idx] * Linv;
}

// DtY[b,k,t] = sum_d D[d,k] * inp[b,d,t]
__global__ __launch_bounds__(256) void dty_kernel(const float* __restrict__ D,
                                                  const float* __restrict__ inp,
                                                  float* __restrict__ DtY) {
  size_t idx = (size_t)blockIdx.x * 256 + threadIdx.x;  // 1,048,576 threads
  int t = (int)(idx % T_DIM);
  size_t rem = idx / T_DIM;
  int kk = (int)(rem % K_DIM);
  int b = (int)(rem / K_DIM);
  const float* ib = inp + (size_t)b * D_DIM * T_DIM + t;
  float acc = 0.f;
  for (int d = 0; d < D_DIM; ++d)
    acc += D[d * K_DIM + kk] * ib[(size_t)d * T_DIM];
  DtY[idx] = acc;
}

// x = x0 ; mx = x0
__global__ __launch_bounds__(256) void init_state_kernel(const float* __restrict__ x0,
                                                         float* __restrict__ x,
                                                         float* __restrict__ mx) {
  size_t stride = (size_t)gridDim.x * 256;
  for (size_t i = (size_t)blockIdx.x * 256 + threadIdx.x; i < N_TOT; i += stride) {
    float v = x0[i];
    x[i] = v;
    mx[i] = v;
  }
}

// zero iteration-indexed accumulators; mom[0]=1, done[0]=0
__global__ __launch_bounds__(1024) void init_scalars_kernel(float* __restrict__ groupsq,
                                                            float* __restrict__ accs,
                                                            float* __restrict__ momArr,
                                                            float* __restrict__ doneArr) {
  int i = threadIdx.x;
  if (i < 8 * MAX_IT) groupsq[i] = 0.f;
  if (i < 2 * MAX_IT) accs[i] = 0.f;
  if (i < MAX_IT + 1) { momArr[i] = 0.f; doneArr[i] = 0.f; }
  if (i == 0) momArr[0] = 1.f;
}

// ---------------------------------------------------------------------------
// K1: Ay = A @ mx (f32 WMMA), fused soft-threshold + per-group sum-of-squares.
// Block: 256 threads = 8 waves; block computes 16 rows x 256 cols; each wave
// owns a 16x32 tile (2 accumulators). A strip and mx K-chunks are staged into
// LDS with async b128 copies (ASYNCcnt), mx double-buffered.
// grid = (T/256=4, K/16=16, B=4). Each block's 16-row strip lies inside one
// 32-row group -> one atomicAdd per block.
// ---------------------------------------------------------------------------
__global__ __launch_bounds__(256)
void fista_gemm_prox_kernel(const float* __restrict__ A,
                            const float* __restrict__ mx,
                            const float* __restrict__ DtY,
                            const float* __restrict__ dLinv,
                            float* __restrict__ l1,
                            float* __restrict__ groupsq,
                            int it) {
  __shared__ __align__(16) float As[16][AS_STRIDE];       // 16 rows x K=256 (+pad)
  __shared__ __align__(16) float Bs[2][16][BS_STRIDE];    // 2 bufs x 16 K-rows x 256 cols (+pad)
  __shared__ float red[256];

  const int tid  = threadIdx.x;
  const int lane = tid & 31;
  const int wave = tid >> 5;
  const int half = lane >> 4;     // 0: lanes 0-15, 1: lanes 16-31
  const int ln   = lane & 15;

  const int i0 = blockIdx.y * 16;                     // row strip
  const int b  = blockIdx.z;                          // batch
  const int t0 = blockIdx.x * 256;                    // block col base
  const int colw = wave * 32;                         // wave col base (local)

  const float* Mb = mx + (size_t)b * K_DIM * T_DIM + t0;

  // ---- stage full A strip (16x256 = 4096 floats = 1024 b128), 4 per thread ----
  const uint32_t asBase = (uint32_t)(uintptr_t)&As[0][0];
#pragma unroll
  for (int j = 0; j < 4; ++j) {
    int q  = tid + 256 * j;          // float4 index, 0..1023
    int r  = q >> 6;                 // row 0..15
    int c4 = q & 63;                 // K/4
    async_copy_b128(asBase + (uint32_t)(r * AS_STRIDE + c4 * 4) * 4u,
                    A + (size_t)(i0 + r) * K_DIM + c4 * 4);
  }
  // ---- stage mx chunk 0 (16 K-rows x 256 cols) ----
  const uint32_t bsBase = (uint32_t)(uintptr_t)&Bs[0][0][0];
  const uint32_t bsBufBytes = (uint32_t)(16 * BS_STRIDE) * 4u;
#pragma unroll
  for (int j = 0; j < 4; ++j) {
    int q  = tid + 256 * j;
    int r  = q >> 6;
    int c4 = q & 63;
    async_copy_b128(bsBase + (uint32_t)(r * BS_STRIDE + c4 * 4) * 4u,
                    Mb + (size_t)r * T_DIM + c4 * 4);
  }

  v8f c0 = {};
  v8f c1 = {};

  int buf = 0;
  for (int cc = 0; cc < 16; ++cc) {            // 16 K-chunks of 16
    if (cc < 15) {
      // prefetch next chunk into the other buffer
      const float* Mc = Mb + (size_t)(cc + 1) * 16 * T_DIM;
      uint32_t dstBase = bsBase + (buf ^ 1) * bsBufBytes;
#pragma unroll
      for (int j = 0; j < 4; ++j) {
        int q  = tid + 256 * j;
        int r  = q >> 6;
        int c4 = q & 63;
        async_copy_b128(dstBase + (uint32_t)(r * BS_STRIDE + c4 * 4) * 4u,
                        Mc + (size_t)r * T_DIM + c4 * 4);
      }
      wait_async_le4();    // in-order completion: current chunk (and A) now resident
    } else {
      wait_async_0();
    }
    __syncthreads();

    const float* Bc = &Bs[buf][0][0];
#pragma unroll
    for (int kl = 0; kl < 16; kl += 4) {
      // A fragment 16x4: M = lane&15, K = 2*half + {0,1}
      v2f a = *(const v2f*)&As[ln][cc * 16 + kl + 2 * half];
      // B fragment 4x16: K rows striped across halves, N = lane&15
      const float* bp = Bc + (kl + 2 * half) * BS_STRIDE + colw + ln;
      v2f b0 = { bp[0],  bp[BS_STRIDE] };
      v2f b1 = { bp[16], bp[BS_STRIDE + 16] };
      c0 = __builtin_amdgcn_wmma_f32_16x16x4_f32(false, a, false, b0,
                                                 (short)0, c0, false, false);
      c1 = __builtin_amdgcn_wmma_f32_16x16x4_f32(false, a, false, b1,
                                                 (short)0, c1, false, false);
    }
    __syncthreads();       // buffer consumed; safe for the prefetch 2 iterations out
    buf ^= 1;
  }

  // Fused epilogue: l1 = max(Ay + DtY/L - lam, 0) + min(Ay + DtY/L + lam, 0)
  const float Linv = *dLinv;
  const float lam  = LAM_C * Linv;
  float ss = 0.f;
#pragma unroll
  for (int r = 0; r < 8; ++r) {
    int row = i0 + r + half * 8;            // C/D layout: rows r / r+8 per half
    size_t g = (size_t)b * K_DIM * T_DIM + (size_t)row * T_DIM + t0 + colw + ln;
    {
      float dy = DtY[g] * Linv;
      float v  = c0[r];
      float lv = fmaxf(v + dy - lam, 0.f) + fminf(v + dy + lam, 0.f);
      l1[g] = lv;
      ss += lv * lv;
    }
    {
      float dy = DtY[g + 16] * Linv;
      float v  = c1[r];
      float lv = fmaxf(v + dy - lam, 0.f) + fminf(v + dy + lam, 0.f);
      l1[g + 16] = lv;
      ss += lv * lv;
    }
  }

  red[tid] = ss;
  __syncthreads();
  for (int s = 128; s > 0; s >>= 1) {
    if (tid < s) red[tid] += red[tid + s];
    __syncthreads();
  }
  if (tid == 0)
    atomicAdd(&groupsq[it * 8 + (i0 >> 5)], red[0]);
}

// ---------------------------------------------------------------------------
// K2: group-L2 prox scale + store new_x, accumulate ||new_x - x||^2, ||x+eps||^2
// ---------------------------------------------------------------------------
__global__ __launch_bounds__(256)
void fista_scale_kernel(const float* __restrict__ l1,
                        const float* __restrict__ x,
                        const float* __restrict__ groupsq,
                        float* __restrict__ newx,
                        float* __restrict__ accs,
                        int it) {
  __shared__ float sscale[8];
  if (threadIdx.x < 8) {
    float nrm = sqrtf(groupsq[it * 8 + threadIdx.x]);
    sscale[threadIdx.x] = (nrm > 0.f) ? fmaxf(0.f, 1.f - REG_C / nrm) : 0.f;
  }
  __syncthreads();

  float dss = 0.f, xss = 0.f;
  size_t stride = (size_t)gridDim.x * 256;
  for (size_t i = (size_t)blockIdx.x * 256 + threadIdx.x; i < N_TOT; i += stride) {
    int row = (int)((i / T_DIM) % K_DIM);
    float nx = l1[i] * sscale[row >> 5];
    float xo = x[i];
    newx[i] = nx;
    float d = nx - xo;
    dss += d * d;
    float xe = xo + 1e-16f;
    xss += xe * xe;
  }

  __shared__ float r1[256];
  __shared__ float r2[256];
  r1[threadIdx.x] = dss;
  r2[threadIdx.x] = xss;
  __syncthreads();
  for (int s = 128; s > 0; s >>= 1) {
    if (threadIdx.x < s) {
      r1[threadIdx.x] += r1[threadIdx.x + s];
      r2[threadIdx.x] += r2[threadIdx.x + s];
    }
    __syncthreads();
  }
  if (threadIdx.x == 0) {
    atomicAdd(&accs[it * 2 + 0], r1[0]);
    atomicAdd(&accs[it * 2 + 1], r2[0]);
  }
}

// ---------------------------------------------------------------------------
// K3: convergence test + momentum commit (freeze logic lives in device data,
// all launches unconditional -> graph-safe and deterministic)
// ---------------------------------------------------------------------------
__global__ __launch_bounds__(256)
void fista_commit_kernel(const float* __restrict__ newx,
                         float* __restrict__ x,
                         float* __restrict__ mx,
                         const float* __restrict__ accs,
                         float* __restrict__ momArr,
                         float* __restrict__ doneArr,
                         int it) {
  float diff = accs[it * 2 + 0];
  float xs   = accs[it * 2 + 1];
  float rel  = sqrtf(diff) / sqrtf(xs);
  bool conv     = rel < TOL_C;
  bool done_old = doneArr[it] != 0.f;
  bool upd      = !(done_old || conv);
  float mom_old = momArr[it];
  float mom_new = 0.5f + 0.5f * sqrtf(1.f + 4.f * mom_old * mom_old);
  float coeff   = (mom_old - 1.f) / mom_new;

  if (blockIdx.x == 0 && threadIdx.x == 0) {
    momArr[it + 1]  = upd ? mom_new : mom_old;
    doneArr[it + 1] = (done_old || conv) ? 1.f : 0.f;
  }
  if (!upd) return;

  size_t stride = (size_t)gridDim.x * 256;
  for (size_t i = (size_t)blockIdx.x * 256 + threadIdx.x; i < N_TOT; i += stride) {
    float xo = x[i];
    float nx = newx[i];
    x[i]  = nx;
    mx[i] = nx + (nx - xo) * coeff;
  }
}

// ---------------------------------------------------------------------------
// Launch
// ---------------------------------------------------------------------------
extern "C" void kernel_launch(void* const* d_in, const int* in_sizes, int n_in,
                              void* d_out, int out_size, void* d_ws, size_t ws_size,
                              hipStream_t stream) {
  const float* D   = (const float*)d_in[0];   // [128,256]
  const float* inp = (const float*)d_in[1];   // [4,128,1024]
  const float* x0  = (const float*)d_in[2];   // [4,256,1024]
  float* x = (float*)d_out;                   // x lives in d_out

  float* ws      = (float*)d_ws;
  float* DtD     = ws;                        // 65536
  float* A       = DtD + 65536;               // 65536
  float* DtY     = A + 65536;                 // 1048576
  float* mxb     = DtY + N_TOT;               // 1048576
  float* l1b     = mxb + N_TOT;               // 1048576
  float* nxb     = l1b + N_TOT;               // 1048576
  float* dLinv   = nxb + N_TOT;               // 1
  float* momArr  = dLinv + 1;                 // 101
  float* doneArr = momArr + (MAX_IT + 1);     // 101
  float* groupsq = doneArr + (MAX_IT + 1);    // 800
  float* accs    = groupsq + 8 * MAX_IT;      // 200

  // ---- one-time setup ----
  dtd_kernel<<<256, 256, 0, stream>>>(D, DtD);
  power_iter_kernel<<<1, 256, 0, stream>>>(DtD, dLinv);
  build_A_kernel<<<256, 256, 0, stream>>>(DtD, dLinv, A);
  dty_kernel<<<4096, 256, 0, stream>>>(D, inp, DtY);
  init_state_kernel<<<1024, 256, 0, stream>>>(x0, x, mxb);
  init_scalars_kernel<<<1, 1024, 0, stream>>>(groupsq, accs, momArr, doneArr);

  // ---- FISTA iterations ----
  dim3 g1(T_DIM / 256, K_DIM / 16, B_DIM);   // (4, 16, 4)
  for (int it = 0; it < MAX_IT; ++it) {
    fista_gemm_prox_kernel<<<g1, 256, 0, stream>>>(A, mxb, DtY, dLinv, l1b, groupsq, it);
    fista_scale_kernel<<<1024, 256, 0, stream>>>(l1b, x, groupsq, nxb, accs, it);
    fista_commit_kernel<<<1024, 256, 0, stream>>>(nxb, x, mxb, accs, momArr, doneArr, it);
  }
}